// FarthestPointSample_7765300871437
// MI455X (gfx1250) — compile-verified
//
#include <hip/hip_runtime.h>
#include <hip/hip_bf16.h>

#define BATCH 16
#define NPTS  16384
#define NSAMP 1024
#define NCH   128
#define BIGF  1e10f

// ---------------------------------------------------------------------------
// Kernel 1: farthest point sampling. One workgroup (32 wave32s) per batch.
// Points and running min-distances live in registers (strided: thread t owns
// points t, t+1024, ..., t+15*1024). Two barriers per iteration.
// ---------------------------------------------------------------------------
__global__ __launch_bounds__(1024) void fps_kernel(const float* __restrict__ p,
                                                   int* __restrict__ idx_out) {
  const int b    = blockIdx.x;
  const int tid  = threadIdx.x;
  const int lane = tid & 31;
  const int wave = tid >> 5;
  const float* __restrict__ pb = p + (size_t)b * NPTS * 3;

  float px[16], py[16], pz[16], mind[16];
#pragma unroll
  for (int s = 0; s < 16; ++s) {
    const int j = tid + (s << 10);
    px[s]   = pb[j * 3 + 0];
    py[s]   = pb[j * 3 + 1];
    pz[s]   = pb[j * 3 + 2];
    mind[s] = BIGF;
  }

  __shared__ float s_red_val[32];
  __shared__ int   s_red_idx[32];
  __shared__ float s_q[3];

  if (tid == 0) {
    idx_out[b * NSAMP] = 0;          // seed index
    s_q[0] = pb[0]; s_q[1] = pb[1]; s_q[2] = pb[2];
  }
  __syncthreads();

  for (int it = 1; it < NSAMP; ++it) {
    const float qx = s_q[0], qy = s_q[1], qz = s_q[2];
    float bestv = -1.0f;
    int   besti = 0;
#pragma unroll
    for (int s = 0; s < 16; ++s) {
      const float dx = px[s] - qx;
      const float dy = py[s] - qy;
      const float dz = pz[s] - qz;
      const float d  = dx * dx + dy * dy + dz * dz;
      const float m  = fminf(mind[s], d);
      mind[s] = m;
      const int j = tid + (s << 10);
      // strictly-greater wins; on exact tie keep the smaller index
      const bool take = (m > bestv) || ((m == bestv) && (j < besti));
      bestv = take ? m : bestv;
      besti = take ? j : besti;
    }
    // wave32 butterfly reduction
#pragma unroll
    for (int off = 16; off > 0; off >>= 1) {
      const float ov = __shfl_xor(bestv, off, 32);
      const int   oi = __shfl_xor(besti, off, 32);
      const bool take = (ov > bestv) || ((ov == bestv) && (oi < besti));
      bestv = take ? ov : bestv;
      besti = take ? oi : besti;
    }
    if (lane == 0) { s_red_val[wave] = bestv; s_red_idx[wave] = besti; }
    __syncthreads();
    if (wave == 0) {
      bestv = s_red_val[lane];
      besti = s_red_idx[lane];
#pragma unroll
      for (int off = 16; off > 0; off >>= 1) {
        const float ov = __shfl_xor(bestv, off, 32);
        const int   oi = __shfl_xor(besti, off, 32);
        const bool take = (ov > bestv) || ((ov == bestv) && (oi < besti));
        bestv = take ? ov : bestv;
        besti = take ? oi : besti;
      }
      if (lane == 0) {
        idx_out[b * NSAMP + it] = besti;
        // winner coords stay hot in L2/SE$ (3MB of points vs 192MB L2)
        s_q[0] = pb[besti * 3 + 0];
        s_q[1] = pb[besti * 3 + 1];
        s_q[2] = pb[besti * 3 + 2];
      }
    }
    __syncthreads();
  }
}

// ---------------------------------------------------------------------------
// Kernel 2: gather. Exactly one wave per output row (2048 blocks x 8 waves =
// 16384 rows). A row of x is 128 f32 = 512 B = 32 lanes x b128, moved with
// the gfx1250 async global<->LDS engine (ASYNCcnt-tracked). The 3-float p_s
// gather is issued between the async load and its wait so it overlaps the
// DMA. The final store's completion is covered by S_ENDPGM's implicit
// wait-idle on all counters.
// ---------------------------------------------------------------------------
__global__ __launch_bounds__(256) void gather_kernel(const float* __restrict__ p,
                                                     const float* __restrict__ x,
                                                     const int* __restrict__ idx,
                                                     float* __restrict__ outP,
                                                     float* __restrict__ outX) {
  __shared__ float lbuf[8][NCH];                 // 512 B per wave
  const int lane = threadIdx.x & 31;
  const int wave = threadIdx.x >> 5;
  const int row  = blockIdx.x * 8 + wave;        // one row per wave
  const int b    = row >> 10;
  const int j    = idx[row];
  // generic LDS address: bits[31:0] are the LDS byte offset (ISA 10.2)
  const unsigned int lds = (unsigned int)(uintptr_t)(&lbuf[wave][lane * 4]);

  const unsigned long long src =
      (unsigned long long)(uintptr_t)(x + ((size_t)b * NPTS + j) * NCH) +
      (unsigned long long)(lane * 16);
  const unsigned long long dst =
      (unsigned long long)(uintptr_t)(outX + (size_t)row * NCH) +
      (unsigned long long)(lane * 16);

  asm volatile("global_load_async_to_lds_b128 %0, %1, off"
               :: "v"(lds), "v"(src) : "memory");

  // overlap the tiny p_s gather with the in-flight async DMA
  if (lane < 3) {
    outP[(size_t)row * 3 + lane] = p[((size_t)b * NPTS + j) * 3 + lane];
  }

  asm volatile("s_wait_asynccnt 0x0" ::: "memory");
  asm volatile("global_store_async_from_lds_b128 %0, %1, off"
               :: "v"(dst), "v"(lds) : "memory");
  // no trailing wait: S_ENDPGM implicitly waits for all counters (ASYNCcnt)
}

// ---------------------------------------------------------------------------
extern "C" void kernel_launch(void* const* d_in, const int* in_sizes, int n_in,
                              void* d_out, int out_size, void* d_ws, size_t ws_size,
                              hipStream_t stream) {
  (void)in_sizes; (void)n_in; (void)out_size; (void)ws_size;
  const float* p = (const float*)d_in[0];   // [16,16384,3]
  const float* x = (const float*)d_in[1];   // [16,16384,128]
  int* idx    = (int*)d_ws;                 // [16,1024] scratch
  float* outP = (float*)d_out;                              // [16,1024,3]
  float* outX = outP + (size_t)BATCH * NSAMP * 3;           // [16,1024,128]

  fps_kernel<<<BATCH, 1024, 0, stream>>>(p, idx);
  gather_kernel<<<(BATCH * NSAMP) / 8, 256, 0, stream>>>(p, x, idx, outP, outX);
}